// TapTracker_48541720379861
// MI455X (gfx1250) — compile-verified
//
#include <hip/hip_runtime.h>

#define HW_TOT   1024
#define NKEYS    40960
#define NCH      64
#define NT       40
#define TOPK_N   10
#define RAD2     144            // dist < 12  <=>  dy*dy+dx*dx < 144 (integer grid)
#define INV_TEMP (1.0f / 0.07f)
#define NWAVES   8

typedef float v2f __attribute__((ext_vector_type(2)));
typedef float v8f __attribute__((ext_vector_type(8)));

// ---------------------------------------------------------------------------
// Pass 1: channel-wise L2 normalize query (with 1/TEMP folded in) and key.
// ---------------------------------------------------------------------------
__global__ __launch_bounds__(256) void tt_normalize(
    const float* __restrict__ q, const float* __restrict__ k,
    float* __restrict__ qn, float* __restrict__ kn) {
  int i = blockIdx.x * blockDim.x + threadIdx.x;
  if (i < HW_TOT) {
    float s = 0.f;
#pragma unroll 8
    for (int c = 0; c < NCH; ++c) { float x = q[c * HW_TOT + i]; s = fmaf(x, x, s); }
    float inv = INV_TEMP / fmaxf(sqrtf(s), 1e-12f);
#pragma unroll 8
    for (int c = 0; c < NCH; ++c) qn[c * HW_TOT + i] = q[c * HW_TOT + i] * inv;
  } else {
    int j = i - HW_TOT;
    if (j < NKEYS) {
      float s = 0.f;
#pragma unroll 8
      for (int c = 0; c < NCH; ++c) { float x = k[c * NKEYS + j]; s = fmaf(x, x, s); }
      float inv = 1.0f / fmaxf(sqrtf(s), 1e-12f);
#pragma unroll 8
      for (int c = 0; c < NCH; ++c) kn[c * NKEYS + j] = k[c * NKEYS + j] * inv;
    }
  }
}

// Running top-K insert with O(1) fast reject against the cached minimum.
__device__ __forceinline__ void topk_insert(float (&tv)[TOPK_N], int (&ti)[TOPK_N],
                                            float& tmin, int& tslot,
                                            float sc, int gid) {
  if (sc > tmin) {
    tv[tslot] = sc;
    ti[tslot] = gid;
    float m = tv[0];
    int ms = 0;
#pragma unroll
    for (int j = 1; j < TOPK_N; ++j) {
      if (tv[j] < m) { m = tv[j]; ms = j; }
    }
    tmin = m;
    tslot = ms;
  }
}

// ---------------------------------------------------------------------------
// Pass 2: fused affinity-GEMM (fp32 WMMA) + circle mask + top-10 + softmax +
// value gather. One block = 16 consecutive queries (all in one image row).
// Software-pipelined: tile N's loads are issued, then tile N-1's mask/top-k
// VALU work runs in the load shadow, then tile N's WMMA chain executes.
// ---------------------------------------------------------------------------
__global__ __launch_bounds__(256) void tt_main(
    const float* __restrict__ qn, const float* __restrict__ kn,
    const float* __restrict__ val, float* __restrict__ out) {
  __shared__ float sVal[NWAVES * 32 * TOPK_N];
  __shared__ int   sIdx[NWAVES * 32 * TOPK_N];
  __shared__ float sW[16 * TOPK_N];
  __shared__ int   sI[16 * TOPK_N];

  const int tid   = threadIdx.x;
  const int wave  = tid >> 5;       // 0..7
  const int lane  = tid & 31;
  const int lhalf = lane >> 4;      // 0: lanes 0-15, 1: lanes 16-31
  const int lmod  = lane & 15;

  const int qbase = blockIdx.x * 16;      // 16 consecutive queries
  const int qy    = qbase >> 5;           // all queries of the tile share a row
  const int qx    = (qbase & 31) + lmod;  // this lane's query column

  // ---- Preload B operands (queries) in ISA B-matrix layout --------------
  // VGPR j of chunk kk holds channel (4*kk + j) for lanes 0-15 and
  // channel (4*kk + j + 2) for lanes 16-31; N = lane%16.
  v2f b[16];
#pragma unroll
  for (int kk = 0; kk < 16; ++kk) {
    int c0 = 4 * kk + 2 * lhalf;
    b[kk].x = qn[c0 * HW_TOT + qbase + lmod];
    b[kk].y = qn[(c0 + 1) * HW_TOT + qbase + lmod];
  }

  // ---- Per-lane running top-10 ------------------------------------------
  float tv[TOPK_N];
  int   ti[TOPK_N];
#pragma unroll
  for (int j = 0; j < TOPK_N; ++j) { tv[j] = -1e30f; ti[j] = -1; }
  float tmin = -1e30f;
  int   tslot = 0;

  // Mask + insert the 8 scores of a completed tile (D layout: VGPR v holds
  // key row v + 8*lhalf, query column lmod).
  auto flush_tile = [&](const v8f& acc, int kb, bool frame0) {
    const int g0  = kb + 8 * lhalf;          // my 8 keys are consecutive
    const int ky  = (g0 >> 5) & 31;
    const int kx0 = g0 & 31;
    const int dy  = ky - qy;
    const int dy2 = dy * dy;
#pragma unroll
    for (int v = 0; v < 8; ++v) {
      int dx = (kx0 + v) - qx;
      if (frame0 || (dy2 + dx * dx < RAD2))
        topk_insert(tv, ti, tmin, tslot, acc[v], g0 + v);
    }
  };

  // Pending (software-pipeline) state.
  v8f pAcc = {};
  int pKb  = -1;
  int pF0  = 0;

  // ---- Sweep key frames; prune rows outside the radius-12 circle --------
  for (int t = 0; t < NT; ++t) {
    int rlo = 0, rhi = 31;
    if (t > 0) {
      rlo = qy - 11; if (rlo < 0)  rlo = 0;
      rhi = qy + 11; if (rhi > 31) rhi = 31;
    }
    const int base   = t * HW_TOT + rlo * 32;
    const int ntiles = (rhi - rlo + 1) * 2;  // 16-key tiles (32 keys per row)

    for (int tile = wave; tile < ntiles; tile += NWAVES) {
      const int kb = base + tile * 16;

      // Prefetch the next tile this wave will touch (GL2-bound).
      if (tile + NWAVES < ntiles)
        __builtin_prefetch(&kn[base + (tile + NWAVES) * 16], 0, 1);

      // Issue ALL A-operand loads for this tile first (pipelined clauses,
      // one wait), into distinct registers.
      v2f a[16];
#pragma unroll
      for (int kk = 0; kk < 16; ++kk) {
        int c0 = 4 * kk + 2 * lhalf;
        a[kk].x = kn[c0 * NKEYS + kb + lmod];       // A: M = lane%16
        a[kk].y = kn[(c0 + 1) * NKEYS + kb + lmod];
      }

      // Previous tile's mask + top-k runs in the shadow of the loads above.
      if (pKb >= 0) flush_tile(pAcc, pKb, pF0 != 0);

      // 16x16 score tile: chain 16 fp32 WMMAs over K = 64 channels.
      v8f acc = {};
#pragma unroll
      for (int kk = 0; kk < 16; ++kk)
        acc = __builtin_amdgcn_wmma_f32_16x16x4_f32(
            false, a[kk], false, b[kk], (short)0, acc, false, false);

      pAcc = acc;
      pKb  = kb;
      pF0  = (t == 0);
    }
  }
  // Drain the pipeline.
  if (pKb >= 0) flush_tile(pAcc, pKb, pF0 != 0);

  // ---- Spill per-lane candidates and merge across waves/halves ----------
  {
    int s = (wave * 32 + lane) * TOPK_N;
#pragma unroll
    for (int j = 0; j < TOPK_N; ++j) { sVal[s + j] = tv[j]; sIdx[s + j] = ti[j]; }
  }
  __syncthreads();

  if (tid < 16) {   // one thread finalizes one query: 160 -> 10, then softmax
    float fv[TOPK_N];
    int   fi[TOPK_N];
#pragma unroll
    for (int j = 0; j < TOPK_N; ++j) { fv[j] = -1e30f; fi[j] = -1; }
    float fmin = -1e30f;
    int   fslot = 0;
    for (int w = 0; w < NWAVES; ++w) {
      for (int h = 0; h < 2; ++h) {
        int s = (w * 32 + h * 16 + tid) * TOPK_N;
#pragma unroll
        for (int j = 0; j < TOPK_N; ++j)
          topk_insert(fv, fi, fmin, fslot, sVal[s + j], sIdx[s + j]);
      }
    }
    float m = fv[0];
#pragma unroll
    for (int j = 1; j < TOPK_N; ++j) m = fmaxf(m, fv[j]);
    float e[TOPK_N];
    float sum = 0.f;
#pragma unroll
    for (int j = 0; j < TOPK_N; ++j) { e[j] = __expf(fv[j] - m); sum += e[j]; }
    float inv = 1.0f / sum;
#pragma unroll
    for (int j = 0; j < TOPK_N; ++j) {
      sW[tid * TOPK_N + j] = e[j] * inv;
      sI[tid * TOPK_N + j] = fi[j];
    }
  }
  __syncthreads();

  // ---- Weighted value gather: thread (c = tid%64) x 4 query groups ------
  const int c = tid & 63;
  for (int qq = tid >> 6; qq < 16; qq += 4) {
    float s = 0.f;
#pragma unroll
    for (int j = 0; j < TOPK_N; ++j)
      s = fmaf(sW[qq * TOPK_N + j], val[c * NKEYS + sI[qq * TOPK_N + j]], s);
    out[c * HW_TOT + qbase + qq] = s;
  }
}

// ---------------------------------------------------------------------------
extern "C" void kernel_launch(void* const* d_in, const int* in_sizes, int n_in,
                              void* d_out, int out_size, void* d_ws, size_t ws_size,
                              hipStream_t stream) {
  const float* q = (const float*)d_in[0];   // (64,1024)
  const float* k = (const float*)d_in[1];   // (64,40960)
  const float* v = (const float*)d_in[2];   // (64,40960)
  // d_in[3] = bool mask — recomputed analytically on device, unused.

  float* qn = (float*)d_ws;                 // 64*1024 floats
  float* kn = qn + NCH * HW_TOT;            // 64*40960 floats (~10.3 MB total)
  float* out = (float*)d_out;               // (64,1024) fp32

  const int total = HW_TOT + NKEYS;                       // 41984
  tt_normalize<<<(total + 255) / 256, 256, 0, stream>>>(q, k, qn, kn);
  tt_main<<<HW_TOT / 16, 256, 0, stream>>>(qn, kn, v, out);
}